// LSJACOBI_6519760355646
// MI455X (gfx1250) — compile-verified
//
#include <hip/hip_runtime.h>
#include <math.h>

typedef float v2f __attribute__((ext_vector_type(2)));
typedef float v8f __attribute__((ext_vector_type(8)));

#define F_IN 256
#define HDIM 128
#define KORD 10
#define OUTD 40
#define OUTP 64          // padded output cols for GEMM2 (4 x 16 tiles)
#define SCAN_B 1024
#define CT 4             // 16-col tiles per wave in GEMM

// ---------------- graph preprocessing ----------------

__global__ void LSJ_init_cnt(int* __restrict__ cnt, int n) {
    int i = blockIdx.x * blockDim.x + threadIdx.x;
    if (i < n) cnt[i] = 1;  // self loop
}

__global__ void LSJ_count_edges(const int* __restrict__ src, const int* __restrict__ dst,
                                int* __restrict__ cnt, int e0) {
    int e = blockIdx.x * blockDim.x + threadIdx.x;
    if (e < e0) {
        atomicAdd(&cnt[src[e]], 1);
        atomicAdd(&cnt[dst[e]], 1);
    }
}

__global__ void LSJ_calc_dinv(const int* __restrict__ cnt, float* __restrict__ dinv, int n) {
    int i = blockIdx.x * blockDim.x + threadIdx.x;
    if (i < n) dinv[i] = 1.0f / sqrtf((float)cnt[i]);
}

__global__ __launch_bounds__(SCAN_B)
void LSJ_scan_block(const int* __restrict__ in, int* __restrict__ out,
                    int* __restrict__ blks, int n) {
    __shared__ int sh[SCAN_B];
    int t = threadIdx.x;
    int i = blockIdx.x * SCAN_B + t;
    int x = (i < n) ? in[i] : 0;
    sh[t] = x;
    __syncthreads();
    for (int d = 1; d < SCAN_B; d <<= 1) {
        int v = (t >= d) ? sh[t - d] : 0;
        __syncthreads();
        sh[t] += v;
        __syncthreads();
    }
    if (i < n) out[i] = sh[t] - x;          // exclusive within block
    if (t == SCAN_B - 1) blks[blockIdx.x] = sh[t];
}

__global__ void LSJ_scan_tot(int* __restrict__ blks, int nb, int* __restrict__ off, int n) {
    if (threadIdx.x == 0 && blockIdx.x == 0) {
        int run = 0;
        for (int b = 0; b < nb; ++b) { int v = blks[b]; blks[b] = run; run += v; }
        off[n] = run;
    }
}

__global__ __launch_bounds__(SCAN_B)
void LSJ_scan_add(int* __restrict__ off, const int* __restrict__ blks, int n) {
    int i = blockIdx.x * SCAN_B + threadIdx.x;
    if (i < n) off[i] += blks[blockIdx.x];
}

__global__ void LSJ_copy_i32(const int* __restrict__ a, int* __restrict__ b, int n) {
    int i = blockIdx.x * blockDim.x + threadIdx.x;
    if (i < n) b[i] = a[i];
}

__global__ void LSJ_fill_edges(const int* __restrict__ src, const int* __restrict__ dst,
                               const float* __restrict__ dinv, int* __restrict__ cursor,
                               int* __restrict__ ecol, float* __restrict__ ew,
                               int e0, int m) {
    int e = blockIdx.x * blockDim.x + threadIdx.x;
    if (e >= m) return;
    int r, c;
    if (e < e0)            { r = src[e];        c = dst[e]; }
    else if (e < 2 * e0)   { r = dst[e - e0];   c = src[e - e0]; }
    else                   { r = e - 2 * e0;    c = r; }
    int pos = atomicAdd(&cursor[r], 1);
    ecol[pos] = c;
    ew[pos] = dinv[r] * dinv[c];
}

// ---------------- weight transpose / padding ----------------
// Bt[n][k] = (n < Nsrc) ? B[k][n] : 0   for n in [0,Npad)
__global__ void LSJ_transpose(const float* __restrict__ B, float* __restrict__ Bt,
                              int K, int Nsrc, int Npad) {
    int idx = blockIdx.x * blockDim.x + threadIdx.x;
    int total = K * Npad;
    if (idx >= total) return;
    int n = idx / K;
    int k = idx - n * K;
    Bt[(size_t)n * K + k] = (n < Nsrc) ? B[(size_t)k * Nsrc + n] : 0.0f;
}

__global__ void LSJ_pad_bias(const float* __restrict__ b, float* __restrict__ bp,
                             int Nsrc, int Npad) {
    int i = blockIdx.x * blockDim.x + threadIdx.x;
    if (i < Npad) bp[i] = (i < Nsrc) ? b[i] : 0.0f;
}

// ---------------- Jacobi coefficients + attention mix ----------------
// satt layout: [(layer*11 + j)*128 + channel]
__global__ __launch_bounds__(128)
void LSJ_coef(const float* __restrict__ att, const float* __restrict__ alphas,
              float* __restrict__ satt) {
    __shared__ float C[KORD + 1][KORD + 1];
    if (threadIdx.x == 0) {
        float al[KORD + 1];
        for (int k = 0; k <= KORD; ++k) al[k] = tanhf(alphas[k]);  // ALPHA = 1
        const float a = 1.0f, b = 1.0f, l = -1.0f, r = 1.0f;
        for (int j = 0; j <= KORD; ++j) C[0][j] = 0.0f;
        C[0][0] = 1.0f;
        {   // L = 1
            float coef1 = ((a - b) * 0.5f - (a + b + 2.0f) * 0.5f * (l + r) / (r - l)) * al[0];
            float coef2 = ((a + b + 2.0f) / (r - l)) * al[0];
            for (int mth = 0; mth <= KORD; ++mth)
                C[1][mth] = coef1 * C[0][mth] + coef2 * ((mth > 0) ? C[0][mth - 1] : 0.0f);
        }
        for (int L = 2; L <= KORD; ++L) {
            float Lf = (float)L;
            float coef_l     = 2.0f * Lf * (Lf + a + b) * (2.0f * Lf - 2.0f + a + b);
            float coef_lm1_1 = (2.0f * Lf + a + b - 1.0f) * (2.0f * Lf + a + b) * (2.0f * Lf + a + b - 2.0f);
            float coef_lm1_2 = (2.0f * Lf + a + b - 1.0f) * (a * a - b * b);
            float coef_lm2   = 2.0f * (Lf - 1.0f + a) * (Lf - 1.0f + b) * (2.0f * Lf + a + b);
            float tmp1  = al[L - 1] * (coef_lm1_1 / coef_l);
            float tmp2  = al[L - 1] * (coef_lm1_2 / coef_l);
            float tmp3  = al[L - 1] * al[L - 2] * (coef_lm2 / coef_l);
            float tmp12 = tmp1 * (2.0f / (r - l));
            float tmp22 = tmp1 * ((r + l) / (r - l)) + tmp2;
            for (int mth = 0; mth <= KORD; ++mth) {
                float s1 = (mth > 0) ? C[L - 1][mth - 1] : 0.0f;
                C[L][mth] = tmp12 * s1 - tmp22 * C[L - 1][mth] - tmp3 * C[L - 2][mth];
            }
        }
    }
    __syncthreads();
    int t = threadIdx.x;  // channel
    for (int i = 0; i < 2; ++i)
        for (int j = 0; j <= KORD; ++j) {
            float s = 0.0f;
            for (int k = 0; k <= KORD; ++k)
                s += att[(i * HDIM + t) * (KORD + 1) + k] * C[k][j];
            satt[(i * (KORD + 1) + j) * HDIM + t] = s;
        }
}

// ---------------- fp32 WMMA GEMM ----------------
// C[M, ldc] (cols colBase..colBase+63) = A[M,Kd] x Bt[64cols][Kd]^T + bias
// One wave computes a 16x64 output strip: 4 accumulators share one A fragment.
// A fragment: VGPR0 = K=0 (lanes0-15) / K=2 (lanes16-31), VGPR1 = K=1 / K=3.
// Bt pre-transposed -> both A and B fragments are contiguous b64 loads.
// Single-stage software pipeline: next K-step's 5 loads issue before the
// current step's 4 WMMAs, so WMMA latency hides load latency (no full
// loadcnt drain per WMMA).
__global__ __launch_bounds__(64)
void LSJ_gemm_wmma(const float* __restrict__ A, const float* __restrict__ Bt,
                   const float* __restrict__ bias, float* __restrict__ C,
                   int Kd, int nColGroups, int totalWaves, int ldc) {
    int waveId = (blockIdx.x * blockDim.x + threadIdx.x) >> 5;
    if (waveId >= totalWaves) return;       // wave-uniform
    int lane = threadIdx.x & 31;
    int rowTile = waveId / nColGroups;
    int colGrp  = waveId - rowTile * nColGroups;
    int rowBase = rowTile << 4;
    int colBase = colGrp * (16 * CT);
    int half = lane >> 4;
    int lr   = lane & 15;

    v8f acc[CT];
#pragma unroll
    for (int t = 0; t < CT; ++t) acc[t] = {};

    const float* ap = A  + (size_t)(rowBase + lr) * Kd + (half << 1);
    const float* bp = Bt + (size_t)(colBase + lr) * Kd + (half << 1);

    // prologue loads
    v2f a_cur = *(const v2f*)ap;
    v2f b_cur[CT];
#pragma unroll
    for (int t = 0; t < CT; ++t)
        b_cur[t] = *(const v2f*)(bp + (size_t)(t * 16) * Kd);

    int k0;
    for (k0 = 0; k0 + 4 < Kd; k0 += 4) {
        // issue next step's loads first (independent of current WMMAs)
        v2f a_nxt = *(const v2f*)(ap + k0 + 4);
        v2f b_nxt[CT];
#pragma unroll
        for (int t = 0; t < CT; ++t)
            b_nxt[t] = *(const v2f*)(bp + (size_t)(t * 16) * Kd + k0 + 4);
        // consume current fragments
#pragma unroll
        for (int t = 0; t < CT; ++t)
            acc[t] = __builtin_amdgcn_wmma_f32_16x16x4_f32(false, a_cur, false, b_cur[t],
                                                           (short)0, acc[t], false, false);
        a_cur = a_nxt;
#pragma unroll
        for (int t = 0; t < CT; ++t) b_cur[t] = b_nxt[t];
    }
    // epilogue: last K-step
#pragma unroll
    for (int t = 0; t < CT; ++t)
        acc[t] = __builtin_amdgcn_wmma_f32_16x16x4_f32(false, a_cur, false, b_cur[t],
                                                       (short)0, acc[t], false, false);

    int row0 = rowBase + (half << 3);       // D: VGPR v -> M=v (l0-15), M=v+8 (l16-31)
#pragma unroll
    for (int t = 0; t < CT; ++t) {
        int col = colBase + t * 16 + lr;
        float bb = bias[col];
#pragma unroll
        for (int v = 0; v < 8; ++v)
            C[(size_t)(row0 + v) * ldc + col] = acc[t][v] + bb;
    }
}

// ---------------- fused gather-SpMM + weighted accumulation ----------------
// nxt[r,:] = sum_k w[k] * cur[col[k],:] ;  agg[r,:] += nxt[r,:] * sattj[:]
// Neighbor (col,w) pairs staged through LDS: coalesced global loads,
// conflict-free ds broadcast inside the FMA loop.
__global__ __launch_bounds__(HDIM)
void LSJ_spmm_fused(const int* __restrict__ off, const int* __restrict__ ecol,
                    const float* __restrict__ ew, const float* __restrict__ cur,
                    float* __restrict__ nxt, float* __restrict__ agg,
                    const float* __restrict__ sattj) {
    __shared__ int   scol[HDIM];
    __shared__ float sw[HDIM];
    int r = blockIdx.x;
    int t = threadIdx.x;
    int s = off[r], e = off[r + 1];
    float acc = 0.0f;
    for (int base = s; base < e; base += HDIM) {
        int nloc = e - base;
        if (nloc > HDIM) nloc = HDIM;
        __syncthreads();
        if (t < nloc) { scol[t] = ecol[base + t]; sw[t] = ew[base + t]; }
        __syncthreads();
        for (int k = 0; k < nloc; ++k)
            acc = fmaf(sw[k], cur[(size_t)scol[k] * HDIM + t], acc);
    }
    size_t o = (size_t)r * HDIM + t;
    nxt[o] = acc;
    agg[o] += acc * sattj[t];
}

__global__ void LSJ_scale_init(const float* __restrict__ cur, float* __restrict__ agg,
                               const float* __restrict__ satt0, size_t n) {
    size_t i = (size_t)blockIdx.x * blockDim.x + threadIdx.x;
    if (i < n) agg[i] = cur[i] * satt0[i & (HDIM - 1)];
}

__global__ void LSJ_relu(const float* __restrict__ in, float* __restrict__ out, size_t n) {
    size_t i = (size_t)blockIdx.x * blockDim.x + threadIdx.x;
    if (i < n) out[i] = fmaxf(in[i], 0.0f);
}

// ---------------- log_softmax over 40 valid cols (stride ldl), one wave per row ----------------
__global__ __launch_bounds__(256)
void LSJ_log_softmax(const float* __restrict__ logits, float* __restrict__ out,
                     int n, int ldl) {
    int wave = threadIdx.x >> 5;
    int lane = threadIdx.x & 31;
    int row = blockIdx.x * 8 + wave;
    if (row >= n) return;                   // wave-uniform
    const float* p = logits + (size_t)row * ldl;
    float v0 = p[lane];
    float v1 = (lane < OUTD - 32) ? p[lane + 32] : -3.402823466e38f;
    float m = fmaxf(v0, v1);
    for (int i = 16; i > 0; i >>= 1) m = fmaxf(m, __shfl_xor(m, i, 32));
    float s = expf(v0 - m) + ((lane < OUTD - 32) ? expf(v1 - m) : 0.0f);
    for (int i = 16; i > 0; i >>= 1) s += __shfl_xor(s, i, 32);
    float ls = logf(s);
    float* q = out + (size_t)row * OUTD;
    q[lane] = v0 - m - ls;
    if (lane < OUTD - 32) q[lane + 32] = v1 - m - ls;
}

// ---------------- host-side orchestration ----------------

extern "C" void kernel_launch(void* const* d_in, const int* in_sizes, int n_in,
                              void* d_out, int out_size, void* d_ws, size_t ws_size,
                              hipStream_t stream) {
    const float* x      = (const float*)d_in[0];
    const int*   src    = (const int*)  d_in[1];
    const int*   dst    = (const int*)  d_in[2];
    const float* W0     = (const float*)d_in[3];
    const float* b0     = (const float*)d_in[4];
    const float* W1     = (const float*)d_in[5];
    const float* b1     = (const float*)d_in[6];
    const float* W2     = (const float*)d_in[7];
    const float* b2     = (const float*)d_in[8];
    const float* att    = (const float*)d_in[9];
    const float* alphas = (const float*)d_in[10];

    const int N  = in_sizes[0] / F_IN;       // 50000
    const int E0 = in_sizes[1];              // 800000
    const int M  = 2 * E0 + N;               // 1,650,000 edge instances
    const size_t NH = (size_t)N * HDIM;

    // workspace carve (256B aligned)
    char* wsb = (char*)d_ws;
    size_t posn = 0;
    auto carve = [&](size_t bytes) -> char* {
        char* p = wsb + posn;
        posn = (posn + bytes + 255) & ~(size_t)255;
        return p;
    };
    int*   cnt    = (int*)  carve((size_t)N * 4);
    int*   offp   = (int*)  carve((size_t)(N + 1) * 4);
    int*   cursor = (int*)  carve((size_t)N * 4);
    int*   blks   = (int*)  carve(4096 * 4);
    float* dinv   = (float*)carve((size_t)N * 4);
    int*   ecol   = (int*)  carve((size_t)M * 4);
    float* ew     = (float*)carve((size_t)M * 4);
    float* satt   = (float*)carve(2 * (KORD + 1) * HDIM * 4);
    float* Bt0    = (float*)carve((size_t)HDIM * F_IN * 4);   // [128][256]
    float* Bt1    = (float*)carve((size_t)HDIM * HDIM * 4);   // [128][128]
    float* Bt2    = (float*)carve((size_t)OUTP * HDIM * 4);   // [64][128] zero-padded
    float* b2p    = (float*)carve(OUTP * 4);
    float* bufA   = (float*)carve(NH * 4);
    float* bufB   = (float*)carve(NH * 4);
    float* bufC   = (float*)carve(NH * 4);

    // ---- CSR build (once per call) ----
    LSJ_init_cnt<<<(N + 255) / 256, 256, 0, stream>>>(cnt, N);
    LSJ_count_edges<<<(E0 + 255) / 256, 256, 0, stream>>>(src, dst, cnt, E0);
    LSJ_calc_dinv<<<(N + 255) / 256, 256, 0, stream>>>(cnt, dinv, N);
    int nb = (N + SCAN_B - 1) / SCAN_B;
    LSJ_scan_block<<<nb, SCAN_B, 0, stream>>>(cnt, offp, blks, N);
    LSJ_scan_tot<<<1, 32, 0, stream>>>(blks, nb, offp, N);
    LSJ_scan_add<<<nb, SCAN_B, 0, stream>>>(offp, blks, N);
    LSJ_copy_i32<<<(N + 255) / 256, 256, 0, stream>>>(offp, cursor, N);
    LSJ_fill_edges<<<(M + 255) / 256, 256, 0, stream>>>(src, dst, dinv, cursor, ecol, ew, E0, M);

    // ---- weight transpose / pad, coefficient mix ----
    {
        int t0 = F_IN * HDIM;
        LSJ_transpose<<<(t0 + 255) / 256, 256, 0, stream>>>(W0, Bt0, F_IN, HDIM, HDIM);
        int t1 = HDIM * HDIM;
        LSJ_transpose<<<(t1 + 255) / 256, 256, 0, stream>>>(W1, Bt1, HDIM, HDIM, HDIM);
        int t2 = HDIM * OUTP;
        LSJ_transpose<<<(t2 + 255) / 256, 256, 0, stream>>>(W2, Bt2, HDIM, OUTD, OUTP);
        LSJ_pad_bias<<<1, OUTP, 0, stream>>>(b2, b2p, OUTD, OUTP);
    }
    LSJ_coef<<<1, 128, 0, stream>>>(att, alphas, satt);

    auto gemm = [&](const float* A, int Kd, const float* Bt, const float* bias,
                    int Nb, float* Cp, int ldc) {
        int nColGroups = Nb / (16 * CT);
        int totalWaves = (N / 16) * nColGroups;
        int blocks = (totalWaves + 1) / 2;   // 2 waves per 64-thread block
        LSJ_gemm_wmma<<<blocks, 64, 0, stream>>>(A, Bt, bias, Cp, Kd, nColGroups,
                                                 totalWaves, ldc);
    };

    int grid_nh = (int)((NH + 255) / 256);

    // ---- layer 0 ----
    gemm(x, F_IN, Bt0, b0, HDIM, bufA, HDIM);
    float* cur = bufA; float* nxt = bufB; float* agg = bufC;
    LSJ_scale_init<<<grid_nh, 256, 0, stream>>>(cur, agg, satt + 0 * HDIM, NH);
    for (int j = 1; j <= KORD; ++j) {
        LSJ_spmm_fused<<<N, HDIM, 0, stream>>>(offp, ecol, ew, cur, nxt, agg, satt + j * HDIM);
        float* t = cur; cur = nxt; nxt = t;
    }
    LSJ_relu<<<grid_nh, 256, 0, stream>>>(agg, cur, NH);     // g -> cur

    // ---- layer 1 ----
    gemm(cur, HDIM, Bt1, b1, HDIM, nxt, HDIM);               // h_lin -> nxt
    { float* t = cur; cur = nxt; nxt = t; }
    LSJ_scale_init<<<grid_nh, 256, 0, stream>>>(cur, agg, satt + (KORD + 1) * HDIM, NH);
    for (int j = 1; j <= KORD; ++j) {
        LSJ_spmm_fused<<<N, HDIM, 0, stream>>>(offp, ecol, ew, cur, nxt, agg,
                                               satt + ((KORD + 1) + j) * HDIM);
        float* t = cur; cur = nxt; nxt = t;
    }
    LSJ_relu<<<grid_nh, 256, 0, stream>>>(agg, cur, NH);     // h2 -> cur

    // ---- output head (padded to 64 cols, stride 64) ----
    float* logits = nxt;                                     // free buffer (N x 64)
    gemm(cur, HDIM, Bt2, b2p, OUTP, logits, OUTP);
    LSJ_log_softmax<<<(N + 7) / 8, 256, 0, stream>>>(logits, (float*)d_out, N, OUTP);
}